// ThreeBodySolver_7868380086676
// MI455X (gfx1250) — compile-verified
//
#include <hip/hip_runtime.h>

#define NN 40000
#define NE 640000
#define IN_DIM 7
#define HID 128
#define NLAYERS 5

typedef __attribute__((ext_vector_type(2))) float v2f;
typedef __attribute__((ext_vector_type(8))) float v8f;

// ---------------- utility: zero fill ----------------
__global__ void zero_f32_kernel(float* __restrict__ p, int n) {
    int i = blockIdx.x * blockDim.x + threadIdx.x;
    if (i < n) p[i] = 0.0f;
}

// ---------------- edge weights + per-dst weight sums ----------------
__global__ void edge_w_kernel(const float* __restrict__ dist,
                              const long long* __restrict__ ei,
                              float* __restrict__ w, float* __restrict__ sw) {
    int e = blockIdx.x * blockDim.x + threadIdx.x;
    if (e >= NE) return;
    float d = dist[e] + 1e-6f;
    float we = 1.0f / (d * d);
    w[e] = we;
    int dst = (int)ei[NE + e];
    unsafeAtomicAdd(&sw[dst], we);
}

// ---------------- encoder: h = x @ enc_W.T + enc_b ----------------
__global__ void encoder_kernel(const float* __restrict__ x,
                               const float* __restrict__ W,
                               const float* __restrict__ b,
                               float* __restrict__ h) {
    __shared__ float xr[IN_DIM];
    int n = blockIdx.x;
    int j = threadIdx.x;           // 0..127
    if (j < IN_DIM) xr[j] = x[n * IN_DIM + j];
    __syncthreads();
    float acc = b[j];
#pragma unroll
    for (int k = 0; k < IN_DIM; ++k) acc += xr[k] * W[j * IN_DIM + k];
    h[n * HID + j] = acc;
}

// ---------------- weighted gather/scatter: agg[dst] += w_e * h[src] ----------------
// one wave (32 lanes x float4) per edge
__global__ void scatter_kernel(const float* __restrict__ h,
                               const long long* __restrict__ ei,
                               const float* __restrict__ w,
                               float* __restrict__ agg) {
    int wave = (blockIdx.x * blockDim.x + threadIdx.x) >> 5;
    int lane = threadIdx.x & 31;
    if (wave >= NE) return;
    int src = (int)ei[wave];
    int dst = (int)ei[NE + wave];
    float we = w[wave];
    const float4 hv = *(const float4*)(h + (size_t)src * HID + lane * 4);
    float* out = agg + (size_t)dst * HID + lane * 4;
    unsafeAtomicAdd(out + 0, we * hv.x);
    unsafeAtomicAdd(out + 1, we * hv.y);
    unsafeAtomicAdd(out + 2, we * hv.z);
    unsafeAtomicAdd(out + 3, we * hv.w);
}

// ---------------- h = relu(agg @ W.T + b * sw) via V_WMMA_F32_16X16X4_F32 ----------------
// 8 waves/block; each wave owns a 16(M) x 128(N) strip (8 C tiles).
// LDS weight layout is K-pair interleaved so each lane's B fragment
// {B[k][n], B[k+1][n]} is one contiguous 8-byte ds_load_b64 into an
// even-aligned VGPR pair (no shuffle movs between ds_load and v_wmma).
//   WtP[p*256 + n*2 + q] = W[n*HID + 2p + q]    (p = k/2, q = k&1)
__global__ __launch_bounds__(256) void gemm_relu_kernel(
        const float* __restrict__ agg, const float* __restrict__ W,
        const float* __restrict__ bias, const float* __restrict__ sw,
        float* __restrict__ h) {
    __shared__ float WtP[HID * HID];          // 64 KB
    int tid = threadIdx.x;
    for (int idx = tid; idx < HID * HID; idx += 256) {
        int p = idx >> 8;                     // K pair index 0..63
        int r = idx & 255;
        int n = r >> 1;                       // output feature 0..127
        int q = r & 1;                        // which K of the pair
        WtP[idx] = W[n * HID + 2 * p + q];
    }
    __syncthreads();

    int wv = tid >> 5, lane = tid & 31;
    int half = lane >> 4;                     // 0: lanes 0-15, 1: lanes 16-31
    int l16 = lane & 15;
    int m0 = blockIdx.x * 128 + wv * 16;

    v8f c[8];
#pragma unroll
    for (int t = 0; t < 8; ++t)
#pragma unroll
        for (int q = 0; q < 8; ++q) c[t][q] = 0.0f;

    // A-fragment source row for this lane (clamped for the partial last block;
    // stores are guarded, so the clamp only affects dead lanes' math)
    int rowA = m0 + l16;
    if (rowA >= NN) rowA = NN - 1;
    const float* arow = agg + (size_t)rowA * HID;

    for (int kb = 0; kb < 32; ++kb) {
        // lanes 0-15 consume K = 4kb,4kb+1 ; lanes 16-31 consume K = 4kb+2,4kb+3
        int k0 = kb * 4 + half * 2;
        v2f a = *(const v2f*)(arow + k0);
        int p = 2 * kb + half;                // K-pair index for the B fragment
        const float* bbase = &WtP[p * 256 + l16 * 2];
#pragma unroll
        for (int t = 0; t < 8; ++t) {
            v2f b = *(const v2f*)(bbase + t * 32);   // {B[k0][n0+l16], B[k0+1][n0+l16]}
            c[t] = __builtin_amdgcn_wmma_f32_16x16x4_f32(
                false, a, false, b, (short)0, c[t], false, false);
        }
    }

    // D layout: VGPR v, lanes 0-15 -> (M=v, N=l16); lanes 16-31 -> (M=v+8, N=l16)
#pragma unroll
    for (int t = 0; t < 8; ++t) {
        int n0 = t * 16;
        float bj = bias[n0 + l16];
#pragma unroll
        for (int v = 0; v < 8; ++v) {
            int m = m0 + v + half * 8;
            if (m < NN) {
                float val = c[t][v] + bj * sw[m];
                h[(size_t)m * HID + n0 + l16] = fmaxf(val, 0.0f);
            }
        }
    }
}

// ---------------- heads: softmax(h @ cW.T + cb) and energy partial sum ----------------
// one wave per node
__global__ void head_kernel(const float* __restrict__ h,
                            const float* __restrict__ cW,
                            const float* __restrict__ cb,
                            const float* __restrict__ eW,
                            float* __restrict__ out,
                            float* __restrict__ esum) {
    int wv = threadIdx.x >> 5, lane = threadIdx.x & 31;
    int n = blockIdx.x * 8 + wv;
    if (n >= NN) return;
    const float4 hv = *(const float4*)(h + (size_t)n * HID + lane * 4);
    float d[4];
#pragma unroll
    for (int r = 0; r < 3; ++r) {
        const float4 wr = *(const float4*)(cW + r * HID + lane * 4);
        d[r] = hv.x * wr.x + hv.y * wr.y + hv.z * wr.z + hv.w * wr.w;
    }
    {
        const float4 we = *(const float4*)(eW + lane * 4);
        d[3] = hv.x * we.x + hv.y * we.y + hv.z * we.z + hv.w * we.w;
    }
#pragma unroll
    for (int r = 0; r < 4; ++r)
#pragma unroll
        for (int off = 16; off > 0; off >>= 1)
            d[r] += __shfl_xor(d[r], off, 32);
    if (lane == 0) {
        float l0 = d[0] + cb[0], l1 = d[1] + cb[1], l2 = d[2] + cb[2];
        float mx = fmaxf(l0, fmaxf(l1, l2));
        float e0 = __expf(l0 - mx), e1 = __expf(l1 - mx), e2 = __expf(l2 - mx);
        float inv = 1.0f / (e0 + e1 + e2);
        out[(size_t)n * 3 + 0] = e0 * inv;
        out[(size_t)n * 3 + 1] = e1 * inv;
        out[(size_t)n * 3 + 2] = e2 * inv;
        unsafeAtomicAdd(esum, d[3]);
    }
}

__global__ void finalize_kernel(const float* __restrict__ esum,
                                const float* __restrict__ eb,
                                float* __restrict__ out) {
    if (threadIdx.x == 0) {
        float e = esum[0] / (float)NN + eb[0];
        out[(size_t)NN * 3 + 0] = e;
        out[(size_t)NN * 3 + 1] = (e < -1.0f) ? 1.0f : 0.0f;
    }
}

extern "C" void kernel_launch(void* const* d_in, const int* in_sizes, int n_in,
                              void* d_out, int out_size, void* d_ws, size_t ws_size,
                              hipStream_t stream) {
    const float*     x      = (const float*)d_in[0];
    const long long* ei     = (const long long*)d_in[1];   // int64 [2, E]
    const float*     dist   = (const float*)d_in[2];
    const float*     enc_W  = (const float*)d_in[3];
    const float*     enc_b  = (const float*)d_in[4];
    const float*     conv_W = (const float*)d_in[5];       // [5,128,128]
    const float*     conv_b = (const float*)d_in[6];       // [5,128]
    const float*     cW     = (const float*)d_in[7];
    const float*     cb     = (const float*)d_in[8];
    const float*     eW     = (const float*)d_in[9];
    const float*     eb     = (const float*)d_in[10];
    float* out = (float*)d_out;

    // workspace layout (floats): w[E] | sw[N] | h[N*H] | agg[N*H] | esum[1]
    float* w    = (float*)d_ws;
    float* sw   = w + NE;
    float* h    = sw + NN;
    float* agg  = h + (size_t)NN * HID;
    float* esum = agg + (size_t)NN * HID;

    zero_f32_kernel<<<(NN + 255) / 256, 256, 0, stream>>>(sw, NN);
    zero_f32_kernel<<<1, 32, 0, stream>>>(esum, 1);
    edge_w_kernel<<<NE / 256, 256, 0, stream>>>(dist, ei, w, sw);
    encoder_kernel<<<NN, HID, 0, stream>>>(x, enc_W, enc_b, h);

    const int aggN = NN * HID;
    for (int l = 0; l < NLAYERS; ++l) {
        zero_f32_kernel<<<(aggN + 255) / 256, 256, 0, stream>>>(agg, aggN);
        scatter_kernel<<<NE / 8, 256, 0, stream>>>(h, ei, w, agg);
        gemm_relu_kernel<<<(NN + 127) / 128, 256, 0, stream>>>(
            agg, conv_W + (size_t)l * HID * HID, conv_b + (size_t)l * HID, sw, h);
    }

    head_kernel<<<(NN + 7) / 8, 256, 0, stream>>>(h, cW, cb, eW, out, esum);
    finalize_kernel<<<1, 32, 0, stream>>>(esum, eb, out);
}